// AbstractSNClustering_83915071030206
// MI455X (gfx1250) — compile-verified
//
#include <hip/hip_runtime.h>

// AbstractSNClustering fused kernel for gfx1250 (MI455X, wave32).
// One wave processes a 16-row tile:
//   stage 1: dist = |c|^2 - 2*(x @ centers^T) via v_wmma_f32_16x16x4_f32,
//            argmin over 64 centers (in-lane over 4 N-tiles, then
//            shfl_xor 1/2/4/8 within each 16-lane half).
//   stage 2: per-row epilogue (16 lanes per row, 2 rows in flight):
//            x_tune = sigmoid(hidden . W_tune + b), folded SN sum, blend.

typedef float v2f __attribute__((ext_vector_type(2)));
typedef float v8f __attribute__((ext_vector_type(8)));

#define DXC 32
#define DSC 16
#define HC  128
#define KC  64
#define NSNC 4

__global__ __launch_bounds__(256)
void snclu_fused_kernel(const float* __restrict__ x,
                        const float* __restrict__ s,
                        const float* __restrict__ hidden,
                        const float* __restrict__ naive_pred,
                        const float* __restrict__ centers,
                        const float* __restrict__ W_tune,
                        const float* __restrict__ b_tune,
                        const float* __restrict__ W_sn,
                        const float* __restrict__ b_sn,
                        const float* __restrict__ rsw,
                        float* __restrict__ out,
                        long nrows)
{
    const int lane = threadIdx.x & 31;
    const int wave = threadIdx.x >> 5;
    const int p    = lane & 15;   // position within 16-lane half
    const int hi   = lane >> 4;   // which half (0/1)
    const int kh   = hi * 2;      // K offset inside each K=4 chunk (ISA A/B f32 layout)

    const long m0 = ((long)blockIdx.x * (blockDim.x >> 5) + (long)wave) * 16;
    if (m0 + 16 > nrows) return;  // uniform per wave: EXEC stays all-ones for WMMA

    // ---- Prefetch hidden rows for the epilogue (global_prefetch_b8) ----
    {
        const float* hrow = hidden + (m0 + p) * HC;
        __builtin_prefetch(hrow + (kh + 0) * 32, 0, 1);
        __builtin_prefetch(hrow + (kh + 1) * 32, 0, 1);
    }

    // ---- Stage 1: A fragments (x tile, 16x32, 8 chunks of K=4) ----
    // A 16x4 f32 layout: VGPR0 = {K0 | K2}, VGPR1 = {K1 | K3} across lane halves.
    v2f a[8];
    {
        const float* xrow = x + (m0 + p) * DXC;
#pragma unroll
        for (int c = 0; c < 8; ++c)
            a[c] = *(const v2f*)(xrow + 4 * c + kh);   // 8-byte aligned
    }

    float bestv[8];
    int   bestn[8];

#pragma unroll
    for (int nt = 0; nt < 4; ++nt) {
        const int n = nt * 16 + p;                       // this lane's center column
        const float* crow = centers + (long)n * DXC;

        v2f bf[8];
        float cn = 0.0f;                                 // half of |center_n|^2
#pragma unroll
        for (int c = 0; c < 8; ++c) {
            bf[c] = *(const v2f*)(crow + 4 * c + kh);    // B[k][n] = centers[n][k]
            cn += bf[c].x * bf[c].x + bf[c].y * bf[c].y;
        }
        cn += __shfl_xor(cn, 16, 32);                    // partner half holds other K's

        v8f acc = {};
#pragma unroll
        for (int c = 0; c < 8; ++c) {
            // 8 args: (neg_a, A, neg_b, B, c_mod, C, reuse_a, reuse_b)
            acc = __builtin_amdgcn_wmma_f32_16x16x4_f32(
                false, a[c], false, bf[c], (short)0, acc, false, false);
        }

        // C/D layout: VGPR r -> row (r + 8*hi), column = p (+16*nt here)
#pragma unroll
        for (int r = 0; r < 8; ++r) {
            const float d = cn - 2.0f * acc[r];
            if (nt == 0 || d < bestv[r] || (d == bestv[r] && n < bestn[r])) {
                bestv[r] = d;
                bestn[r] = n;
            }
        }
    }

    // Cross-lane argmin within each 16-lane half (rows r in half0, r+8 in half1).
#pragma unroll
    for (int r = 0; r < 8; ++r) {
        float bv = bestv[r];
        int   bn = bestn[r];
#pragma unroll
        for (int m = 1; m <= 8; m <<= 1) {
            const float ov = __shfl_xor(bv, m, 32);
            const int   on = __shfl_xor(bn, m, 32);
            if (ov < bv || (ov == bv && on < bn)) { bv = ov; bn = on; }
        }
        bestn[r] = bn;
    }

    // ---- Stage 2: per-row epilogue, 16 lanes cooperate per row ----
    const float bt = b_tune[0];
    float wt[8];                                          // W_tune chunk for lane p
    {
        const float* w = W_tune + p * 8;
#pragma unroll
        for (int j = 0; j < 8; ++j) wt[j] = w[j];
    }

#pragma unroll
    for (int r = 0; r < 8; ++r) {
        const long row  = m0 + (long)hi * 8 + r;          // half0: rows 0..7, half1: 8..15
        const int  cidx = bestn[r];                       // uniform within each half

        // Folded SN term: sum_n w_n*(W_sn[n,cidx,:].s + b_sn[n,cidx])
        float wsum = 0.0f, bsum = 0.0f;
#pragma unroll
        for (int n = 0; n < NSNC; ++n) {
            const float wn = rsw[cidx * NSNC + n];
            wsum += wn * W_sn[(n * KC + cidx) * DSC + p];
            bsum += wn * b_sn[n * KC + cidx];
        }
        float part_sn = s[row * DSC + p] * wsum;

        // Tune term: lane p dots hidden[row, 8p .. 8p+7] (coalesced 512B row read)
        const float4* h4 = (const float4*)(hidden + row * HC + p * 8);
        float part_t = 0.0f;
        {
            const float4 h0 = h4[0], h1 = h4[1];
            part_t += h0.x * wt[0] + h0.y * wt[1] + h0.z * wt[2] + h0.w * wt[3];
            part_t += h1.x * wt[4] + h1.y * wt[5] + h1.z * wt[6] + h1.w * wt[7];
        }

        // One packed 16-lane reduction for both partials.
#pragma unroll
        for (int m = 1; m <= 8; m <<= 1) {
            part_sn += __shfl_xor(part_sn, m, 32);
            part_t  += __shfl_xor(part_t,  m, 32);
        }

        if (p == 0) {
            const float x_sn_sum = part_sn + bsum;
            const float t  = 1.0f / (1.0f + __expf(-(part_t + bt)));
            const float np = naive_pred[row];
            out[row] = x_sn_sum + t * (np - x_sn_sum);
        }
    }
}

extern "C" void kernel_launch(void* const* d_in, const int* in_sizes, int n_in,
                              void* d_out, int out_size, void* d_ws, size_t ws_size,
                              hipStream_t stream) {
    const float* x          = (const float*)d_in[0];
    const float* s          = (const float*)d_in[1];
    const float* hidden     = (const float*)d_in[2];
    const float* naive_pred = (const float*)d_in[3];
    const float* centers    = (const float*)d_in[4];
    const float* W_tune     = (const float*)d_in[5];
    const float* b_tune     = (const float*)d_in[6];
    const float* W_sn       = (const float*)d_in[7];
    const float* b_sn       = (const float*)d_in[8];
    const float* rsw        = (const float*)d_in[9];
    float* out              = (float*)d_out;

    const long B = (long)in_sizes[0] / DXC;               // rows
    const int  block = 256;                               // 8 waves = 128 rows/block
    const long rowsPerBlock = (block / 32) * 16;
    const long grid = (B + rowsPerBlock - 1) / rowsPerBlock;

    snclu_fused_kernel<<<dim3((unsigned)grid), dim3(block), 0, stream>>>(
        x, s, hidden, naive_pred, centers, W_tune, b_tune, W_sn, b_sn, rsw, out, B);
}